// HCTA_58729382805594
// MI455X (gfx1250) — compile-verified
//
#include <hip/hip_runtime.h>
#include <hip/hip_bf16.h>
#include <math.h>

#define NN        2048
#define C_INN     16
#define TT        8
#define VVV       21
#define MMM       2
#define VMM       42                    // V*M
#define COUT      64
#define NUM_NODES (NN*TT*VMM)           // 688128
#define E_INCC    (4*NUM_NODES)         // 2752512
#define EMBED     336
#define HEADS     4
#define HEAD_DIM  84
#define ROWS      (NN*TT)               // 16384
#define OUTCOLS   (8*EMBED)             // 2688
#define NODE_STRIDE 21504               // 64*336 = per-n stride of (node,64) flat

typedef __attribute__((ext_vector_type(2))) float v2f;
typedef __attribute__((ext_vector_type(8))) float v8f;
typedef unsigned int u32x4 __attribute__((ext_vector_type(4)));
typedef int          i32x4 __attribute__((ext_vector_type(4)));
typedef int          i32x8 __attribute__((ext_vector_type(8)));

// ---------------- BatchNorm statistics ----------------
__global__ void k_bn_stats(const float* __restrict__ x, float* __restrict__ acc) {
    int blk = blockIdx.x;               // n*16 + c
    int c = blk & 15;
    const float* p = x + (size_t)blk * 336;
    float s = 0.f, s2 = 0.f;
    for (int i = threadIdx.x; i < 336; i += blockDim.x) {
        float v = p[i];
        s += v; s2 += v * v;
    }
    __shared__ float sh0[256], sh1[256];
    sh0[threadIdx.x] = s; sh1[threadIdx.x] = s2;
    __syncthreads();
    for (int st = 128; st > 0; st >>= 1) {
        if (threadIdx.x < st) {
            sh0[threadIdx.x] += sh0[threadIdx.x + st];
            sh1[threadIdx.x] += sh1[threadIdx.x + st];
        }
        __syncthreads();
    }
    if (threadIdx.x == 0) {
        atomicAdd(&acc[c], sh0[0]);
        atomicAdd(&acc[16 + c], sh1[0]);
    }
}

__global__ void k_bn_final(const float* __restrict__ acc,
                           const float* __restrict__ gamma,
                           const float* __restrict__ beta,
                           float* __restrict__ ss) {
    int c = threadIdx.x;
    if (c < 16) {
        const float cnt = (float)NUM_NODES;
        float mean = acc[c] / cnt;
        float var  = acc[16 + c] / cnt - mean * mean;
        float rs   = rsqrtf(var + 1e-5f);
        float sc   = gamma[c] * rs;
        ss[c]      = sc;
        ss[16 + c] = beta[c] - mean * sc;
    }
}

// ---------------- 16 -> 64 projection (res path and hypergraph xt) ----------------
__global__ void k_proj16(const float* __restrict__ x, const float* __restrict__ w,
                         const float* __restrict__ bias, const float* __restrict__ ss,
                         float* __restrict__ out) {
    long gid = (long)blockIdx.x * blockDim.x + threadIdx.x;  // NUM_NODES*64
    int o = (int)(gid & 63);
    long node = gid >> 6;
    int n = (int)(node / 336);
    int p = (int)(node % 336);
    const float* xp = x + (size_t)n * 5376 + p;              // stride 336 per channel
    float s = bias ? bias[o] : 0.0f;
#pragma unroll
    for (int c = 0; c < 16; ++c) {
        float xv = xp[(size_t)c * 336];
        if (ss) xv = xv * ss[c] + ss[16 + c];
        s += w[o * 16 + c] * xv;
    }
    out[gid] = s;
}

// ---------------- degrees ----------------
__global__ void k_degree(const int* __restrict__ ni, const int* __restrict__ ei,
                         float* __restrict__ D, float* __restrict__ B) {
    long i = (long)blockIdx.x * blockDim.x + threadIdx.x;
    if (i < E_INCC) {
        atomicAdd(&D[ni[i]], 1.0f);
        atomicAdd(&B[ei[i]], 1.0f);
    }
}

__global__ void k_recip(float* __restrict__ d) {
    long i = (long)blockIdx.x * blockDim.x + threadIdx.x;
    float v = d[i];
    d[i] = (v > 0.f) ? (1.0f / v) : 0.0f;
}

// ---------------- scatter-add:  dst[to[i]][c] += w(from[i]) * src[from[i]][c] ----------------
__global__ void k_scatter(const float* __restrict__ src, const int* __restrict__ from,
                          const int* __restrict__ to, const float* __restrict__ wsrc,
                          float* __restrict__ dst) {
    long gid = (long)blockIdx.x * blockDim.x + threadIdx.x;  // E_INC * 16
    long i = gid >> 4;
    int cg = (int)(gid & 15);
    int f = from[i];
    int t = to[i];
    float wgt = wsrc ? wsrc[f] : 1.0f;
    const float4 v = *(const float4*)(src + (size_t)f * 64 + cg * 4);
    float* d = dst + (size_t)t * 64 + cg * 4;
    atomicAdd(d + 0, wgt * v.x);
    atomicAdd(d + 1, wgt * v.y);
    atomicAdd(d + 2, wgt * v.z);
    atomicAdd(d + 3, wgt * v.w);
}

__global__ void k_hfinal(float* __restrict__ h, const float* __restrict__ dinv,
                         const float* __restrict__ hcb) {
    long gid = (long)blockIdx.x * blockDim.x + threadIdx.x;  // NUM_NODES*64
    h[gid] = h[gid] * dinv[gid >> 6] + hcb[gid & 63];
}

// ---------------- seq: T-mix conv + reinterpret/transpose ----------------
__global__ void k_seq(const float* __restrict__ h, const float* __restrict__ cw,
                      const float* __restrict__ cb, float* __restrict__ seq) {
    long gid = (long)blockIdx.x * blockDim.x + threadIdx.x;  // ROWS*EMBED
    int n  = (int)(gid / 2688);
    int r  = (int)(gid % 2688);
    int s  = r / 336;
    int e2 = r % 336;
    int w  = e2 / 42;
    int vm = e2 % 42;
    const float* hp = h + (size_t)n * NODE_STRIDE + (size_t)(8 * w) * 336 + vm;
    float acc = cb[s];
#pragma unroll
    for (int t = 0; t < 8; ++t) acc += cw[s * 8 + t] * hp[t * 42];
    seq[gid] = acc;
}

// ---------------- TDM: stage one 16 x K fp32 tile of A into LDS ----------------
// D# per CDNA5 ISA: 2D tile, data_size=4B, tile_dim0=K, tile_dim1=16,
// tensor_dim0=K (stride K), tensor_dim1=R, groups 2/3 zero (<=2D tensor).
__device__ __forceinline__ void tdm_load_tile_a(const float* A, int tileM, int K, int R,
                                                unsigned lds_off) {
#if defined(__has_builtin) && __has_builtin(__builtin_amdgcn_tensor_load_to_lds)
    unsigned long long ga = (unsigned long long)(uintptr_t)A +
                            (unsigned long long)tileM * 16ull * (unsigned)K * 4ull;
    u32x4 g0;
    g0[0] = 1u;                                              // count=1, user descriptor
    g0[1] = lds_off;                                         // lds_addr (bytes)
    g0[2] = (unsigned)ga;                                    // global_addr[31:0]
    g0[3] = (unsigned)((ga >> 32) & 0x01FFFFFFull) | (2u << 30); // ga[56:32] | type=2
    i32x8 g1;
    g1[0] = (int)(2u << 16);                                 // workgroup_mask=0, data_size=4B
    g1[1] = (int)(((unsigned)K & 0xFFFFu) << 16);            // tensor_dim0[15:0] @ bits 63:48
    g1[2] = (int)((((unsigned)K >> 16) & 0xFFFFu) |          // tensor_dim0[31:16]
                  (((unsigned)R & 0xFFFFu) << 16));          // tensor_dim1[15:0]
    g1[3] = (int)((((unsigned)R >> 16) & 0xFFFFu) |          // tensor_dim1[31:16]
                  (((unsigned)K & 0xFFFFu) << 16));          // tile_dim0 = K
    g1[4] = 16;                                              // tile_dim1=16, tile_dim2=0
    g1[5] = (int)(unsigned)K;                                // tensor_dim0_stride[31:0]
    g1[6] = 0;                                               // stride hi, dim1_stride lo
    g1[7] = 0;
    i32x4 z4 = {0, 0, 0, 0};
#if __clang_major__ >= 23
    i32x8 z8 = {0, 0, 0, 0, 0, 0, 0, 0};
    __builtin_amdgcn_tensor_load_to_lds(g0, g1, z4, z4, z8, 0);
#else
    __builtin_amdgcn_tensor_load_to_lds(g0, g1, z4, z4, 0);
#endif
    __builtin_amdgcn_s_wait_tensorcnt(0);
#endif
}

// ---------------- fp32 WMMA GEMM with TDM-staged A tile ----------------
// Block = 4 waves: one 16-row M tile, 4 adjacent 16-col N tiles. A staged in LDS by TDM
// (wave 0), read via ds_load; W fragments from global (L2-resident weights).
template <int K, int MODE>
__global__ void k_gemm_wmma(const float* __restrict__ A, const float* __restrict__ W,
                            const float* __restrict__ bias, const float* __restrict__ resid,
                            float* __restrict__ C, int R, int Ncol) {
    __shared__ float As[16 * K];
    int tileM = blockIdx.x;
    int tileN = blockIdx.y * 4 + threadIdx.y;
    int lane = threadIdx.x;
    int half = lane >> 4;
    int l16  = lane & 15;

#if defined(__has_builtin) && __has_builtin(__builtin_amdgcn_tensor_load_to_lds)
    if (threadIdx.y == 0) {
        tdm_load_tile_a(A, tileM, K, R, (unsigned)(uintptr_t)As);
    }
#else
    {   // cooperative fallback: 16*K floats, float4 chunks
        const float4* src = (const float4*)(A + (size_t)tileM * 16 * K);
        for (int i = threadIdx.y * 32 + threadIdx.x; i < 16 * K / 4; i += 128)
            ((float4*)As)[i] = src[i];
    }
#endif
    __syncthreads();

    if (tileN * 16 < Ncol) {                 // wave-uniform; after barrier, safe
        const float* Ar = &As[l16 * K];      // LDS row for this lane's M
        const float* Wr = W + (size_t)(tileN * 16 + l16) * K;
        v8f acc = {};
#pragma unroll 4
        for (int k = 0; k < K; k += 4) {
            v2f a, b;
            a[0] = Ar[k + half * 2 + 0];
            a[1] = Ar[k + half * 2 + 1];
            b[0] = Wr[k + half * 2 + 0];
            b[1] = Wr[k + half * 2 + 1];
            acc = __builtin_amdgcn_wmma_f32_16x16x4_f32(
                false, a, false, b, (short)0, acc, false, false);
        }
        int col = tileN * 16 + l16;
        float bval = bias[col];
#pragma unroll
        for (int j = 0; j < 8; ++j) {
            int row = tileM * 16 + half * 8 + j;
            size_t idx = (size_t)row * Ncol + col;
            float v = acc[j] + bval;
            if (MODE) { v += resid[idx]; v = fmaxf(v, 0.0f); }
            C[idx] = v;
        }
    }
}

// ---------------- attention: scores + softmax ----------------
__global__ void k_attn_scores(const float* __restrict__ qq, const float* __restrict__ kk,
                              float* __restrict__ attn) {
    int nh = blockIdx.x;                 // n*4 + h
    int n = nh >> 2, h = nh & 3;
    int tid = threadIdx.x;               // 64 threads = (s,t)
    int s = tid >> 3, t = tid & 7;
    const float* qp = qq + (size_t)(n * 8 + s) * EMBED + h * HEAD_DIM;
    const float* kp = kk + (size_t)(n * 8 + t) * EMBED + h * HEAD_DIM;
    float sc = 0.f;
    for (int d = 0; d < HEAD_DIM; ++d) sc += qp[d] * kp[d];
    sc *= 0.10910894511799618f;          // 1/sqrt(84)
    __shared__ float sh[64];
    sh[tid] = sc;
    __syncthreads();
    float m = -INFINITY;
    for (int j = 0; j < 8; ++j) m = fmaxf(m, sh[s * 8 + j]);
    float e = __expf(sc - m);
    __syncthreads();
    sh[tid] = e;
    __syncthreads();
    float sum = 0.f;
    for (int j = 0; j < 8; ++j) sum += sh[s * 8 + j];
    attn[(size_t)nh * 64 + tid] = e / sum;
}

__global__ void k_attn_ao(const float* __restrict__ attn, const float* __restrict__ vv,
                          float* __restrict__ ao) {
    long gid = (long)blockIdx.x * blockDim.x + threadIdx.x;  // ROWS*EMBED
    int r = (int)(gid / EMBED);          // n*8+s
    int j = (int)(gid % EMBED);
    int n = r >> 3, s = r & 7;
    int h = j / HEAD_DIM;
    const float* ap = attn + ((size_t)(n * 4 + h) * 8 + s) * 8;
    float acc = 0.f;
#pragma unroll
    for (int t = 0; t < 8; ++t)
        acc += ap[t] * vv[(size_t)(n * 8 + t) * EMBED + j];
    ao[gid] = acc;
}

// =====================================================================
extern "C" void kernel_launch(void* const* d_in, const int* in_sizes, int n_in,
                              void* d_out, int out_size, void* d_ws, size_t ws_size,
                              hipStream_t stream) {
    const float* x          = (const float*)d_in[0];
    const int*   hi         = (const int*)  d_in[1];
    const float* hc_w       = (const float*)d_in[2];
    const float* hc_b       = (const float*)d_in[3];
    const float* conv_w     = (const float*)d_in[4];
    const float* conv_b     = (const float*)d_in[5];
    const float* bn_gamma   = (const float*)d_in[6];
    const float* bn_beta    = (const float*)d_in[7];
    const float* proj_w     = (const float*)d_in[8];
    const float* proj_b     = (const float*)d_in[9];
    const float* wq         = (const float*)d_in[10];
    const float* bq         = (const float*)d_in[11];
    const float* wk         = (const float*)d_in[12];
    const float* bk         = (const float*)d_in[13];
    const float* wv         = (const float*)d_in[14];
    const float* bv         = (const float*)d_in[15];
    const float* in_proj_w  = (const float*)d_in[16];
    const float* in_proj_b  = (const float*)d_in[17];
    const float* out_proj_w = (const float*)d_in[18];
    const float* out_proj_b = (const float*)d_in[19];
    const float* fc_w       = (const float*)d_in[20];
    const float* fc_b       = (const float*)d_in[21];
    const int* node_idx = hi;
    const int* edge_idx = hi + E_INCC;
    float* out = (float*)d_out;

    char* w8 = (char*)d_ws;
    size_t off = 0;
    auto alloc = [&](size_t nfloats) -> float* {
        float* p = (float*)(w8 + off);
        off += ((nfloats * sizeof(float) + 255) & ~(size_t)255);
        return p;
    };
    float* bn_acc = alloc(32);
    float* bn_ss  = alloc(32);
    float* res    = alloc((size_t)ROWS * OUTCOLS);        // residual (176 MB)
    float* bufA   = alloc((size_t)NUM_NODES * 64);        // xt, then h
    float* bufB   = alloc((size_t)NUM_NODES * 64);        // e accumulator
    float* Dv     = alloc(NUM_NODES);
    float* Bv     = alloc(NUM_NODES);
    float* S0     = alloc((size_t)ROWS * EMBED);          // seq -> qq -> ao2
    float* S1     = alloc((size_t)ROWS * EMBED);          // q   -> kk
    float* S2     = alloc((size_t)ROWS * EMBED);          // k   -> vv
    float* S3     = alloc((size_t)ROWS * EMBED);          // v   -> ao
    float* attn   = alloc((size_t)NN * HEADS * 64);

    // ---- BatchNorm + residual projection ----
    hipMemsetAsync(bn_acc, 0, 32 * sizeof(float), stream);
    k_bn_stats<<<NN * C_INN, 256, 0, stream>>>(x, bn_acc);
    k_bn_final<<<1, 32, 0, stream>>>(bn_acc, bn_gamma, bn_beta, bn_ss);
    k_proj16<<<(NUM_NODES * 64) / 256, 256, 0, stream>>>(x, proj_w, proj_b, bn_ss, res);

    // ---- Hypergraph conv ----
    k_proj16<<<(NUM_NODES * 64) / 256, 256, 0, stream>>>(x, hc_w, nullptr, nullptr, bufA);
    hipMemsetAsync(Dv, 0, (size_t)NUM_NODES * sizeof(float), stream);
    hipMemsetAsync(Bv, 0, (size_t)NUM_NODES * sizeof(float), stream);
    k_degree<<<E_INCC / 256, 256, 0, stream>>>(node_idx, edge_idx, Dv, Bv);
    k_recip<<<NUM_NODES / 256, 256, 0, stream>>>(Dv);
    k_recip<<<NUM_NODES / 256, 256, 0, stream>>>(Bv);
    hipMemsetAsync(bufB, 0, (size_t)NUM_NODES * 64 * sizeof(float), stream);
    k_scatter<<<(E_INCC * 16) / 256, 256, 0, stream>>>(bufA, node_idx, edge_idx, nullptr, bufB);
    hipMemsetAsync(bufA, 0, (size_t)NUM_NODES * 64 * sizeof(float), stream);
    k_scatter<<<(E_INCC * 16) / 256, 256, 0, stream>>>(bufB, edge_idx, node_idx, Bv, bufA);
    k_hfinal<<<(NUM_NODES * 64) / 256, 256, 0, stream>>>(bufA, Dv, hc_b);

    // ---- seq (T-mix conv) ----
    k_seq<<<(ROWS * EMBED) / 256, 256, 0, stream>>>(bufA, conv_w, conv_b, S0);

    // ---- dense GEMMs on WMMA (TDM-staged A) ----
    dim3 blk(32, 4);
    dim3 g336(ROWS / 16, (EMBED / 16 + 3) / 4);      // (1024, 6)
    dim3 g2688(ROWS / 16, (OUTCOLS / 16 + 3) / 4);   // (1024, 42)
    k_gemm_wmma<EMBED, 0><<<g336, blk, 0, stream>>>(S0, wq, bq, nullptr, S1, ROWS, EMBED);
    k_gemm_wmma<EMBED, 0><<<g336, blk, 0, stream>>>(S0, wk, bk, nullptr, S2, ROWS, EMBED);
    k_gemm_wmma<EMBED, 0><<<g336, blk, 0, stream>>>(S0, wv, bv, nullptr, S3, ROWS, EMBED);
    k_gemm_wmma<EMBED, 0><<<g336, blk, 0, stream>>>(S1, in_proj_w, in_proj_b, nullptr, S0, ROWS, EMBED);                             // qq
    k_gemm_wmma<EMBED, 0><<<g336, blk, 0, stream>>>(S2, in_proj_w + 1 * EMBED * EMBED, in_proj_b + EMBED, nullptr, S1, ROWS, EMBED); // kk
    k_gemm_wmma<EMBED, 0><<<g336, blk, 0, stream>>>(S3, in_proj_w + 2 * EMBED * EMBED, in_proj_b + 2 * EMBED, nullptr, S2, ROWS, EMBED); // vv

    // ---- attention ----
    k_attn_scores<<<NN * HEADS, 64, 0, stream>>>(S0, S1, attn);
    k_attn_ao<<<(ROWS * EMBED) / 256, 256, 0, stream>>>(attn, S2, S3);

    // ---- out_proj, then fc fused with residual + ReLU into d_out ----
    k_gemm_wmma<EMBED, 0><<<g336, blk, 0, stream>>>(S3, out_proj_w, out_proj_b, nullptr, S0, ROWS, EMBED);
    k_gemm_wmma<EMBED, 1><<<g2688, blk, 0, stream>>>(S0, fc_w, fc_b, res, out, ROWS, EMBED);
}